// ResD1Layer_6176162972426
// MI455X (gfx1250) — compile-verified
//
#include <hip/hip_runtime.h>
#include <hip/hip_bf16.h>
#include <math.h>

typedef float v2f __attribute__((ext_vector_type(2)));
typedef float v8f __attribute__((ext_vector_type(8)));

#define B_SZ 1024
#define D_SZ 128
#define H_SZ 128
#define LN_EPS 1e-5f
#define TAU_MIN 0.01f

// ---------------------------------------------------------------------------
// K1: s[i,d] = sum_j |x[i,d] - x[j,d]|   (dominant: B^2*D VALU work)
// 256 blocks x 128 threads; block covers 4 rows i; j tiled 128-wide in LDS.
// Tile fill is float4 (b128 global load + b128 ds store, conflict-free).
// ---------------------------------------------------------------------------
__global__ __launch_bounds__(128) void pairwise_abs_sum(
    const float* __restrict__ x, float* __restrict__ s) {
  __shared__ float tile[128 * 128];   // 64 KB
  const int d  = threadIdx.x;
  const int i0 = blockIdx.x * 4;

  float xi[4], acc[4];
#pragma unroll
  for (int r = 0; r < 4; ++r) {
    xi[r]  = x[(i0 + r) * D_SZ + d];
    acc[r] = 0.0f;
  }

  const float4* __restrict__ x4 = (const float4*)x;
  float4* tile4 = (float4*)tile;

  for (int jb = 0; jb < B_SZ; jb += 128) {
    __syncthreads();
    // 128x128 tile = 4096 float4; 128 threads x 32 iterations
    const int base4 = jb * (D_SZ / 4);   // float4 index of tile start in x
#pragma unroll 8
    for (int k = 0; k < 32; ++k)
      tile4[k * 128 + d] = x4[base4 + k * 128 + d];
    __syncthreads();
#pragma unroll 8
    for (int jj = 0; jj < 128; ++jj) {
      const float v = tile[jj * D_SZ + d];
#pragma unroll
      for (int r = 0; r < 4; ++r) acc[r] += fabsf(xi[r] - v);
    }
  }

#pragma unroll
  for (int r = 0; r < 4; ++r) s[(i0 + r) * D_SZ + d] = acc[r];
}

// ---------------------------------------------------------------------------
// K2: itau[i] = 1 / max(softplus(x[i] . W_tau + b_tau), TAU_MIN)
// Reciprocal computed once per row (exact IEEE div) so the GEMM1 epilogue is
// a multiply instead of a per-element v_div_scale/v_rcp/Newton chain.
// ---------------------------------------------------------------------------
__global__ __launch_bounds__(128) void tau_kernel(
    const float* __restrict__ x, const float* __restrict__ Wt,
    const float* __restrict__ bt, float* __restrict__ itau) {
  __shared__ float red[128];
  const int row = blockIdx.x;
  const int t   = threadIdx.x;
  red[t] = x[row * D_SZ + t] * Wt[t];
  __syncthreads();
  for (int off = 64; off > 0; off >>= 1) {
    if (t < off) red[t] += red[t + off];
    __syncthreads();
  }
  if (t == 0) {
    const float z  = red[0] + bt[0];
    const float sp = (z > 20.0f) ? z : log1pf(expf(z));
    itau[row] = 1.0f / fmaxf(sp, TAU_MIN);
  }
}

// ---------------------------------------------------------------------------
// K3: h_sum = (s @ W_diff^T + B*b_diff) * itau    via V_WMMA_F32_16X16X4_F32
// 64 blocks x 8 waves; wave w owns the 16x16 tile (rows i0.., cols 16w..).
// ---------------------------------------------------------------------------
__global__ __launch_bounds__(256) void gemm1_wmma(
    const float* __restrict__ s, const float* __restrict__ Wd,
    const float* __restrict__ bd, const float* __restrict__ itau,
    float* __restrict__ hsum) {
  const int wave = threadIdx.x >> 5;
  const int lane = threadIdx.x & 31;
  const int lo   = lane & 15;
  const int hi   = lane >> 4;
  const int i0   = blockIdx.x * 16;
  const int n0   = wave * 16;

  v8f acc = {};
  const float* arow = s  + (i0 + lo) * D_SZ + 2 * hi;   // A: 16x4 f32 tile
  const float* brow = Wd + (n0 + lo) * D_SZ + 2 * hi;   // B: W_diff^T tile

#pragma unroll 4
  for (int k = 0; k < D_SZ; k += 4) {
    v2f a = *(const v2f*)(arow + k);
    v2f b = *(const v2f*)(brow + k);
    acc = __builtin_amdgcn_wmma_f32_16x16x4_f32(
        false, a, false, b, (short)0, acc, false, false);
  }

  // D layout: VGPR r -> row (r + 8*hi), col lo
  const float bn = (float)B_SZ * bd[n0 + lo];
#pragma unroll
  for (int r = 0; r < 8; ++r) {
    const int row = i0 + r + 8 * hi;
    hsum[row * H_SZ + n0 + lo] = (acc[r] + bn) * itau[row];
  }
}

// ---------------------------------------------------------------------------
// K4: h = h_sum @ W_agg^T + b_agg; y = h + x; out = LN(y)*gamma + beta
// WMMA into a 16x128 LDS stripe; LN with 16 threads/row + shfl_xor reduce.
// ---------------------------------------------------------------------------
__global__ __launch_bounds__(256) void gemm2_ln(
    const float* __restrict__ hsum, const float* __restrict__ Wa,
    const float* __restrict__ ba, const float* __restrict__ x,
    const float* __restrict__ gamma, const float* __restrict__ beta,
    float* __restrict__ out) {
  __shared__ float y[16][H_SZ + 1];   // +1 pad vs bank conflicts
  const int wave = threadIdx.x >> 5;
  const int lane = threadIdx.x & 31;
  const int lo   = lane & 15;
  const int hi   = lane >> 4;
  const int i0   = blockIdx.x * 16;
  const int n0   = wave * 16;

  v8f acc = {};
  const float* arow = hsum + (i0 + lo) * H_SZ + 2 * hi;
  const float* brow = Wa   + (n0 + lo) * H_SZ + 2 * hi;

#pragma unroll 4
  for (int k = 0; k < H_SZ; k += 4) {
    v2f a = *(const v2f*)(arow + k);
    v2f b = *(const v2f*)(brow + k);
    acc = __builtin_amdgcn_wmma_f32_16x16x4_f32(
        false, a, false, b, (short)0, acc, false, false);
  }

  const int   n  = n0 + lo;
  const float bn = ba[n];
#pragma unroll
  for (int r = 0; r < 8; ++r) {
    const int m = r + 8 * hi;
    y[m][n] = acc[r] + bn + x[(i0 + m) * D_SZ + n];   // residual (H == D)
  }
  __syncthreads();

  // LayerNorm: 16 threads per row (row = tid>>4), shfl_xor within 16 lanes.
  const int rr = threadIdx.x >> 4;   // 0..15
  const int cc = threadIdx.x & 15;   // 0..15

  float p = 0.0f;
  for (int c = cc; c < H_SZ; c += 16) p += y[rr][c];
#pragma unroll
  for (int m = 8; m > 0; m >>= 1) p += __shfl_xor(p, m, 16);
  const float mu = p * (1.0f / H_SZ);

  float q = 0.0f;
  for (int c = cc; c < H_SZ; c += 16) {
    const float dv = y[rr][c] - mu;
    q += dv * dv;
  }
#pragma unroll
  for (int m = 8; m > 0; m >>= 1) q += __shfl_xor(q, m, 16);
  const float inv = rsqrtf(q * (1.0f / H_SZ) + LN_EPS);

  for (int c = cc; c < H_SZ; c += 16)
    out[(i0 + rr) * H_SZ + c] = (y[rr][c] - mu) * inv * gamma[c] + beta[c];
}

// ---------------------------------------------------------------------------
extern "C" void kernel_launch(void* const* d_in, const int* in_sizes, int n_in,
                              void* d_out, int out_size, void* d_ws, size_t ws_size,
                              hipStream_t stream) {
  const float* x     = (const float*)d_in[0];
  const float* Wd    = (const float*)d_in[1];
  const float* bd    = (const float*)d_in[2];
  const float* Wt    = (const float*)d_in[3];
  const float* bt    = (const float*)d_in[4];
  const float* Wa    = (const float*)d_in[5];
  const float* ba    = (const float*)d_in[6];
  const float* gamma = (const float*)d_in[7];
  const float* beta  = (const float*)d_in[8];
  float* out = (float*)d_out;

  float* s    = (float*)d_ws;                 // 1024*128 f32 = 512 KB
  float* hsum = s + B_SZ * D_SZ;              // 512 KB
  float* itau = hsum + B_SZ * H_SZ;           // 4 KB

  pairwise_abs_sum<<<B_SZ / 4, 128, 0, stream>>>(x, s);
  tau_kernel      <<<B_SZ,     128, 0, stream>>>(x, Wt, bt, itau);
  gemm1_wmma      <<<B_SZ / 16, 256, 0, stream>>>(s, Wd, bd, itau, hsum);
  gemm2_ln        <<<B_SZ / 16, 256, 0, stream>>>(hsum, Wa, ba, x, gamma, beta, out);
}